// Model_6425271075031
// MI455X (gfx1250) — compile-verified
//
#include <hip/hip_runtime.h>

// Problem constants (match reference)
#define Bn 128   // batch
#define Sn 128   // STATE
#define An 32    // ACT
#define Nn 160   // nodes = STATE+ACT
#define En 5120  // edges
#define Hn 256   // HID

typedef float v2f __attribute__((ext_vector_type(2)));
typedef float v8f __attribute__((ext_vector_type(8)));

// ---------------------------------------------------------------- prep kernels

__global__ void k_zero(float* __restrict__ p, int n) {
    int i = blockIdx.x * blockDim.x + threadIdx.x;
    if (i < n) p[i] = 0.0f;
}

__global__ void k_concat(const float* __restrict__ state,
                         const float* __restrict__ action,
                         float* __restrict__ x) {
    int i = blockIdx.x * blockDim.x + threadIdx.x;
    if (i < Bn * Nn) {
        int b = i / Nn, j = i % Nn;
        x[i] = (j < Sn) ? state[b * Sn + j] : action[b * An + (j - Sn)];
    }
}

// M[d,s] += 1 per edge (duplicates accumulate), A[d,s] = 1 (set), deg[d] += 1
__global__ void k_edges(const int* __restrict__ ei,
                        float* __restrict__ M, float* __restrict__ A,
                        float* __restrict__ deg) {
    int e = blockIdx.x * blockDim.x + threadIdx.x;
    if (e < En) {
        int s = ei[e];       // edge_index[0] = src
        int d = ei[En + e];  // edge_index[1] = dst
        atomicAdd(&M[d * Nn + s], 1.0f);
        A[d * Nn + s] = 1.0f;
        atomicAdd(&deg[d], 1.0f);
    }
}

// Mn = M / max(deg,1);  also set A diagonal to 1
__global__ void k_norm(const float* __restrict__ M, const float* __restrict__ deg,
                       float* __restrict__ Mn, float* __restrict__ A) {
    int i = blockIdx.x * blockDim.x + threadIdx.x;
    if (i < Nn * Nn) {
        float dg = deg[i / Nn];
        dg = dg < 1.0f ? 1.0f : dg;
        Mn[i] = M[i] / dg;
    }
    if (i < Nn) A[i * Nn + i] = 1.0f;
}

// ------------------------------------------------- fp32 WMMA GEMM (templated)
// C(MD x ND) = A(MD x KD) @ B(KD x ND) [+bias][+resid][relu]
// One wave computes an (MB*16 x NB*16) block of 16x16 tiles:
//   - each A fragment (b64 load) is reused across NB column tiles
//   - each B fragment (2x b32, strided) is reused across MB row tiles
// All MB A fragments are loaded into distinct registers BEFORE the WMMA burst
// so load latency pipelines across the burst instead of s_wait_loadcnt 0.
template <int MD, int ND, int KD, int MB, int NB, bool BIAS, bool RELU, bool RESID>
__global__ void k_gemm(const float* __restrict__ A, const float* __restrict__ Bm,
                       const float* __restrict__ bias, const float* __restrict__ resid,
                       float* __restrict__ C) {
    constexpr int NTT = ND / 16 / NB;
    constexpr int MT  = MD / 16 / MB;
    constexpr int numTiles = MT * NTT;
    const int lane = threadIdx.x & 31;
    const int wid  = threadIdx.x >> 5;
    const int tile = blockIdx.x * (blockDim.x >> 5) + wid;
    if (tile >= numTiles) return;             // wave-uniform: EXEC stays full
    const int tm = tile / NTT, tn = tile % NTT;
    const int half = lane >> 4, lrow = lane & 15;
    const int col0 = tn * (16 * NB) + lrow;
    const float* arow[MB];
#pragma unroll
    for (int mi = 0; mi < MB; ++mi)
        arow[mi] = A + (size_t)((tm * MB + mi) * 16 + lrow) * KD;
    const float* bcol = Bm + col0;
    v8f acc[MB][NB];
#pragma unroll
    for (int mi = 0; mi < MB; ++mi)
#pragma unroll
        for (int ni = 0; ni < NB; ++ni) acc[mi][ni] = {};
    for (int k0 = 0; k0 < KD; k0 += 4) {
        const int k = k0 + half * 2;
        v2f bv[NB];
#pragma unroll
        for (int ni = 0; ni < NB; ++ni) {
            bv[ni].x = bcol[(size_t)k * ND + ni * 16];
            bv[ni].y = bcol[(size_t)(k + 1) * ND + ni * 16];
        }
        v2f av[MB];
#pragma unroll
        for (int mi = 0; mi < MB; ++mi)
            av[mi] = *(const v2f*)(arow[mi] + k);  // 8B-aligned: KD%4==0, k even
#pragma unroll
        for (int mi = 0; mi < MB; ++mi)
#pragma unroll
            for (int ni = 0; ni < NB; ++ni)
                acc[mi][ni] = __builtin_amdgcn_wmma_f32_16x16x4_f32(
                    false, av[mi], false, bv[ni], (short)0, acc[mi][ni], false, false);
    }
#pragma unroll
    for (int mi = 0; mi < MB; ++mi) {
#pragma unroll
        for (int ni = 0; ni < NB; ++ni) {
            const int c = col0 + ni * 16;
#pragma unroll
            for (int g = 0; g < 8; ++g) {
                const int m = (tm * MB + mi) * 16 + g + half * 8;
                float v = acc[mi][ni][g];
                if constexpr (BIAS)  v += bias[c];
                if constexpr (RESID) v += resid[(size_t)m * ND + c];
                if constexpr (RELU)  v = v > 0.0f ? v : 0.0f;
                C[(size_t)m * ND + c] = v;
            }
        }
    }
}

// ------------------------- layer-1 GEMM with fused A-construction (x[b]⊙P[n])
// Rows r = n*128 + b (node-major) so layer-2 aggregation is one big GEMM.
template <int MB, int NB>
__global__ void k_gemm1(const float* __restrict__ x, const float* __restrict__ P,
                        const float* __restrict__ W1, const float* __restrict__ b1,
                        float* __restrict__ h1) {
    constexpr int NTT = Hn / 16 / NB;                 // 8
    constexpr int MT  = (Nn * Bn) / 16 / MB;          // 320
    constexpr int numTiles = MT * NTT;                // 2560
    const int lane = threadIdx.x & 31;
    const int wid  = threadIdx.x >> 5;
    const int tile = blockIdx.x * (blockDim.x >> 5) + wid;
    if (tile >= numTiles) return;
    const int tm = tile / NTT, tn = tile % NTT;
    const int half = lane >> 4, lrow = lane & 15;
    const int col0 = tn * (16 * NB) + lrow;
    const float* xr[MB];
    const float* pr[MB];
#pragma unroll
    for (int mi = 0; mi < MB; ++mi) {
        const int r = (tm * MB + mi) * 16 + lrow;     // r = n*128 + b
        xr[mi] = x + (r & 127) * Nn;                  // b row of x
        pr[mi] = P + (r >> 7) * Nn;                   // n row of P
    }
    const float* bcol = W1 + col0;
    v8f acc[MB][NB];
#pragma unroll
    for (int mi = 0; mi < MB; ++mi)
#pragma unroll
        for (int ni = 0; ni < NB; ++ni) acc[mi][ni] = {};
    for (int k0 = 0; k0 < Nn; k0 += 4) {
        const int k = k0 + half * 2;
        v2f bv[NB];
#pragma unroll
        for (int ni = 0; ni < NB; ++ni) {
            bv[ni].x = bcol[k * Hn + ni * 16];
            bv[ni].y = bcol[(k + 1) * Hn + ni * 16];
        }
        v2f av[MB];
#pragma unroll
        for (int mi = 0; mi < MB; ++mi) {
            v2f xv = *(const v2f*)(xr[mi] + k);
            v2f pv = *(const v2f*)(pr[mi] + k);
            av[mi] = xv * pv;                          // fused x ⊙ P
        }
#pragma unroll
        for (int mi = 0; mi < MB; ++mi)
#pragma unroll
            for (int ni = 0; ni < NB; ++ni)
                acc[mi][ni] = __builtin_amdgcn_wmma_f32_16x16x4_f32(
                    false, av[mi], false, bv[ni], (short)0, acc[mi][ni], false, false);
    }
#pragma unroll
    for (int mi = 0; mi < MB; ++mi) {
#pragma unroll
        for (int ni = 0; ni < NB; ++ni) {
            const int c = col0 + ni * 16;
            const float bias = b1[c];
#pragma unroll
            for (int g = 0; g < 8; ++g) {
                const int m = (tm * MB + mi) * 16 + g + half * 8;
                float v = acc[mi][ni][g] + bias;
                v = v > 0.0f ? v : 0.0f;
                h1[(size_t)m * Hn + c] = v;            // [n][b][f] layout
            }
        }
    }
}

// pooled[b,f] = mean over n of h2[(n*128+b)*256+f]
__global__ void k_pool(const float* __restrict__ h2, float* __restrict__ pooled) {
    int i = blockIdx.x * blockDim.x + threadIdx.x;     // i = b*256 + f
    if (i < Bn * Hn) {
        float s = 0.0f;
        for (int n = 0; n < Nn; ++n) s += h2[(size_t)n * (Bn * Hn) + i];
        pooled[i] = s * (1.0f / (float)Nn);
    }
}

// ---------------------------------------------------------------------- launch

extern "C" void kernel_launch(void* const* d_in, const int* in_sizes, int n_in,
                              void* d_out, int out_size, void* d_ws, size_t ws_size,
                              hipStream_t stream) {
    (void)in_sizes; (void)n_in; (void)out_size; (void)ws_size;
    const float* state  = (const float*)d_in[0];
    const float* action = (const float*)d_in[1];
    const int*   ei     = (const int*)d_in[2];
    const float* W1 = (const float*)d_in[3];
    const float* b1 = (const float*)d_in[4];
    const float* W2 = (const float*)d_in[5];
    const float* b2 = (const float*)d_in[6];
    const float* Wo = (const float*)d_in[7];
    const float* bo = (const float*)d_in[8];
    float* out = (float*)d_out;
    float* ws  = (float*)d_ws;

    // workspace layout (floats)
    float* Mbuf   = ws;                 // 25600  (zeroed)
    float* Abuf   = Mbuf + 25600;       // 25600  (zeroed)
    float* degb   = Abuf + 25600;       // 160    (zeroed) -- contiguous zero region
    float* Mn     = degb + 160;         // 25600
    float* Pb     = Mn   + 25600;       // 25600
    float* xb     = Pb   + 25600;       // 20480
    float* pooled = xb   + 20480;       // 32768
    float* h1     = pooled + 32768;     // 5242880  [n][b][f]
    float* agg2   = h1   + 5242880;     // 5242880  (h2 reuses h1)

    // graph prep
    k_zero  <<<(51360 + 255) / 256, 256, 0, stream>>>(Mbuf, 51360);
    k_concat<<<(Bn * Nn + 255) / 256, 256, 0, stream>>>(state, action, xb);
    k_edges <<<(En + 255) / 256, 256, 0, stream>>>(ei, Mbuf, Abuf, degb);
    k_norm  <<<(Nn * Nn + 255) / 256, 256, 0, stream>>>(Mbuf, degb, Mn, Abuf);

    // P = Mn @ A   (160x160, K=160): MB=5,NB=2 -> 2x5 = 10 tiles -> 3 blocks
    k_gemm<Nn, Nn, Nn, 5, 2, false, false, false>
        <<<3, 128, 0, stream>>>(Mn, Abuf, nullptr, nullptr, Pb);

    // h1 = relu((x⊙P) @ W1 + b1)  (20480x256, K=160): MB=4,NB=2 -> 2560 tiles
    k_gemm1<4, 2><<<640, 128, 0, stream>>>(xb, Pb, W1, b1, h1);

    // agg2 = Mn @ h1  (160x32768, K=160): MB=5,NB=2 -> 2x1024 = 2048 tiles
    k_gemm<Nn, Bn * Hn, Nn, 5, 2, false, false, false>
        <<<512, 128, 0, stream>>>(Mn, h1, nullptr, nullptr, agg2);

    // h2 = relu(agg2 @ W2 + b2)  (20480x256, K=256): MB=4,NB=2 -> 2560 tiles
    k_gemm<Nn * Bn, Hn, Hn, 4, 2, true, true, false>
        <<<640, 128, 0, stream>>>(agg2, W2, b2, nullptr, h1);

    // mean-pool over nodes
    k_pool<<<(Bn * Hn + 255) / 256, 256, 0, stream>>>(h1, pooled);

    // out = pooled @ Wo + bo + state  (128x128, K=256): MB=4,NB=2 -> 8 tiles
    k_gemm<Bn, Sn, Hn, 4, 2, true, false, true>
        <<<2, 128, 0, stream>>>(pooled, Wo, bo, state, out);
}